// CGWeight_90915867722548
// MI455X (gfx1250) — compile-verified
//
#include <hip/hip_runtime.h>

typedef __attribute__((ext_vector_type(2))) float v2f;
typedef __attribute__((ext_vector_type(4))) float v4f;
typedef __attribute__((ext_vector_type(8))) float v8f;

#define KPAD 12      // padded K (j2 dim 9 -> 12)
#define NPAD 32      // padded N (j1*3+k dim 27 -> 32)
#define LROW 36      // LDS row stride in floats (mult of 4 -> 16B aligned cols, odd*4 -> conflict-free rows)

// ---------------- setup: fold weight*cg into padded B matrix Bmat[KPAD][NPAD] ----------------
__global__ void build_bmat(const float* __restrict__ w,
                           const float* __restrict__ cg01, const float* __restrict__ cg10,
                           const float* __restrict__ cg11, const float* __restrict__ cg12,
                           const float* __restrict__ cg21, const float* __restrict__ cg22,
                           float* __restrict__ bmat) {
    int t = threadIdx.x;
    if (t >= KPAD * NPAD) return;
    int kk = t / NPAD;   // j2 index (K)
    int n  = t % NPAD;   // j1*3 + k (N)
    float v = 0.0f;
    if (kk < 9 && n < 27) {
        int j2 = kk, j1 = n / 3, k = n % 3;
        int l1 = (j1 == 0) ? 0 : (j1 < 4 ? 1 : 2);
        int l2 = (j2 == 0) ? 0 : (j2 < 4 ? 1 : 2);
        int j1loc = j1 - ((l1 == 0) ? 0 : (l1 == 1 ? 1 : 4));
        int j2loc = j2 - ((l2 == 0) ? 0 : (l2 == 1 ? 1 : 4));
        int idx = -1;
        const float* cg = nullptr;
        if      (l1 == 0 && l2 == 1) { idx = 0; cg = cg01; }
        else if (l1 == 1 && l2 == 0) { idx = 1; cg = cg10; }
        else if (l1 == 1 && l2 == 1) { idx = 2; cg = cg11; }
        else if (l1 == 1 && l2 == 2) { idx = 3; cg = cg12; }
        else if (l1 == 2 && l2 == 1) { idx = 4; cg = cg21; }
        else if (l1 == 2 && l2 == 2) { idx = 5; cg = cg22; }
        if (idx >= 0) {
            int d2 = 2 * l2 + 1;
            v = w[idx] * cg[(j1loc * d2 + j2loc) * 3 + k];
        }
    }
    bmat[t] = v;
}

// ---------------- main WMMA kernel: 32 edges per wave-chunk ----------------
__global__ void __launch_bounds__(256)
cg_wmma_kernel(const float* __restrict__ a0, const float* __restrict__ h0,
               const float* __restrict__ a1, const float* __restrict__ h1,
               const float* __restrict__ a2, const float* __restrict__ h2,
               const float* __restrict__ bmat, float* __restrict__ out,
               int nChunks) {
    __shared__ alignas(16) float lds[8][32 * LROW];   // per-wave slice, 4608B each

    const int lane = threadIdx.x & 31;
    const int wid  = threadIdx.x >> 5;
    const int ln   = lane & 15;
    const bool lg  = (lane >= 16);          // lane group (K half-split)
    float* gl = &lds[wid][0];

    // Loop-invariant B fragments: B[k, n], lane holds k = s*4 + lg*2 + v, n = t*16 + ln
    v2f bf[3][2];
#pragma unroll
    for (int s = 0; s < 3; ++s)
#pragma unroll
        for (int t = 0; t < 2; ++t) {
            int kb = s * 4 + (lg ? 2 : 0);
            bf[s][t].x = bmat[(kb + 0) * NPAD + t * 16 + ln];
            bf[s][t].y = bmat[(kb + 1) * NPAD + t * 16 + ln];
        }

    const int waveId = wid + blockIdx.x * (blockDim.x >> 5);
    const int nWaves = gridDim.x * (blockDim.x >> 5);

    for (int c = waveId; c < nChunks; c += nWaves) {       // wave-uniform loop: EXEC stays all-ones
        const long base = (long)c * 32;

        // ---- stage 1: two 16-edge subtiles, G = H9 @ Cc via v_wmma_f32_16x16x4_f32 ----
#pragma unroll
        for (int st = 0; st < 2; ++st) {
            const long e = base + st * 16 + ln;           // A-matrix row = lane&15 for both groups
            float hq[12];
            hq[0] = h0[e];
            hq[1] = h1[e * 3 + 0]; hq[2] = h1[e * 3 + 1]; hq[3] = h1[e * 3 + 2];
            hq[4] = h2[e * 5 + 0]; hq[5] = h2[e * 5 + 1]; hq[6] = h2[e * 5 + 2];
            hq[7] = h2[e * 5 + 3]; hq[8] = h2[e * 5 + 4];
            hq[9] = 0.0f; hq[10] = 0.0f; hq[11] = 0.0f;    // K padding

            v8f acc[2] = { {0,0,0,0,0,0,0,0}, {0,0,0,0,0,0,0,0} };
#pragma unroll
            for (int s = 0; s < 3; ++s) {
                v2f af;
                af.x = lg ? hq[s * 4 + 2] : hq[s * 4 + 0];
                af.y = lg ? hq[s * 4 + 3] : hq[s * 4 + 1];
#pragma unroll
                for (int t = 0; t < 2; ++t)
                    acc[t] = __builtin_amdgcn_wmma_f32_16x16x4_f32(
                        false, af, false, bf[s][t], (short)0, acc[t], false, false);
            }
            // D layout: vgpr r, this lane holds (edge = lg*8 + r, n = t*16 + ln)
#pragma unroll
            for (int t = 0; t < 2; ++t) {
                float* p = gl + (t * 16 + ln) * LROW + st * 16 + (lg ? 8 : 0);
                v4f lo = { acc[t][0], acc[t][1], acc[t][2], acc[t][3] };
                v4f hi = { acc[t][4], acc[t][5], acc[t][6], acc[t][7] };
                *(v4f*)(p)     = lo;
                *(v4f*)(p + 4) = hi;
            }
        }

        // ---- stage 2: each lane finishes one edge: out = A9 . G (same-wave LDS, in-order) ----
        const long e2 = base + lane;
        float aq[9];
        aq[0] = a0[e2];
        aq[1] = a1[e2 * 3 + 0]; aq[2] = a1[e2 * 3 + 1]; aq[3] = a1[e2 * 3 + 2];
        aq[4] = a2[e2 * 5 + 0]; aq[5] = a2[e2 * 5 + 1]; aq[6] = a2[e2 * 5 + 2];
        aq[7] = a2[e2 * 5 + 3]; aq[8] = a2[e2 * 5 + 4];

        float o0 = 0.0f, o1 = 0.0f, o2 = 0.0f;
#pragma unroll
        for (int j1 = 0; j1 < 9; ++j1) {
            float g0 = gl[(j1 * 3 + 0) * LROW + lane];
            float g1 = gl[(j1 * 3 + 1) * LROW + lane];
            float g2 = gl[(j1 * 3 + 2) * LROW + lane];
            o0 += aq[j1] * g0;
            o1 += aq[j1] * g1;
            o2 += aq[j1] * g2;
        }
        out[e2 * 3 + 0] = o0;
        out[e2 * 3 + 1] = o1;
        out[e2 * 3 + 2] = o2;
    }
}

// ---------------- generic tail (E % 32 != 0; unused for E = 3.2M) ----------------
__global__ void cg_tail_kernel(const float* __restrict__ a0, const float* __restrict__ h0,
                               const float* __restrict__ a1, const float* __restrict__ h1,
                               const float* __restrict__ a2, const float* __restrict__ h2,
                               const float* __restrict__ bmat, float* __restrict__ out,
                               long start, long E) {
    long e = start + (long)blockIdx.x * blockDim.x + threadIdx.x;
    if (e >= E) return;
    float aq[9], hq[9];
    aq[0] = a0[e];
    aq[1] = a1[e*3+0]; aq[2] = a1[e*3+1]; aq[3] = a1[e*3+2];
    aq[4] = a2[e*5+0]; aq[5] = a2[e*5+1]; aq[6] = a2[e*5+2]; aq[7] = a2[e*5+3]; aq[8] = a2[e*5+4];
    hq[0] = h0[e];
    hq[1] = h1[e*3+0]; hq[2] = h1[e*3+1]; hq[3] = h1[e*3+2];
    hq[4] = h2[e*5+0]; hq[5] = h2[e*5+1]; hq[6] = h2[e*5+2]; hq[7] = h2[e*5+3]; hq[8] = h2[e*5+4];
    float o[3] = {0.f, 0.f, 0.f};
    for (int j1 = 0; j1 < 9; ++j1)
        for (int k = 0; k < 3; ++k) {
            float g = 0.f;
            for (int j2 = 0; j2 < 9; ++j2) g += hq[j2] * bmat[j2 * NPAD + j1 * 3 + k];
            o[k] += aq[j1] * g;
        }
    out[e*3+0] = o[0]; out[e*3+1] = o[1]; out[e*3+2] = o[2];
}

extern "C" void kernel_launch(void* const* d_in, const int* in_sizes, int n_in,
                              void* d_out, int out_size, void* d_ws, size_t ws_size,
                              hipStream_t stream) {
    // setup_inputs() dict order: a0,h0,a1,h1,a2,h2,weight,cg01,cg10,cg11,cg12,cg21,cg22
    const float* a0   = (const float*)d_in[0];
    const float* h0   = (const float*)d_in[1];
    const float* a1   = (const float*)d_in[2];
    const float* h1   = (const float*)d_in[3];
    const float* a2   = (const float*)d_in[4];
    const float* h2   = (const float*)d_in[5];
    const float* w    = (const float*)d_in[6];
    const float* cg01 = (const float*)d_in[7];
    const float* cg10 = (const float*)d_in[8];
    const float* cg11 = (const float*)d_in[9];
    const float* cg12 = (const float*)d_in[10];
    const float* cg21 = (const float*)d_in[11];
    const float* cg22 = (const float*)d_in[12];
    float* out  = (float*)d_out;
    float* bmat = (float*)d_ws;   // KPAD*NPAD = 384 floats

    const long E = in_sizes[0];

    build_bmat<<<1, KPAD * NPAD, 0, stream>>>(w, cg01, cg10, cg11, cg12, cg21, cg22, bmat);

    const int nChunks = (int)(E / 32);
    if (nChunks > 0) {
        int blocks = (nChunks + 7) / 8;          // 8 waves per 256-thread block
        if (blocks > 3125) blocks = 3125;        // ~25k waves -> a few chunks each
        cg_wmma_kernel<<<blocks, 256, 0, stream>>>(a0, h0, a1, h1, a2, h2, bmat, out, nChunks);
    }
    const long rem = E - (long)nChunks * 32;
    if (rem > 0) {
        int blocks = (int)((rem + 255) / 256);
        cg_tail_kernel<<<blocks, 256, 0, stream>>>(a0, h0, a1, h1, a2, h2, bmat, out,
                                                   (long)nChunks * 32, E);
    }
}